// ShiftInvariantCrossEntropy_2963527434282
// MI455X (gfx1250) — compile-verified
//
#include <hip/hip_runtime.h>
#include <math.h>

// ln(300/20)/300, double-accurate then truncated to float
#define LOG_BIN_WIDTH_F 0.0090268340036740336f
#define NUM_C 300
#define BLK_ROWS 8   // 8 waves of 32 = 256 threads per block, one row per wave

typedef __attribute__((ext_vector_type(2))) float v2f;
typedef __attribute__((ext_vector_type(8))) float v8f;

__device__ __forceinline__ float wave_sum32(float v) {
#pragma unroll
    for (int off = 16; off > 0; off >>= 1) v += __shfl_xor(v, off, 32);
    return v;
}
__device__ __forceinline__ float wave_max32(float v) {
#pragma unroll
    for (int off = 16; off > 0; off >>= 1) v = fmaxf(v, __shfl_xor(v, off, 32));
    return v;
}

// Stage 1: one wave32 per row. Lane handles k = c*128 + 4*lane + j,
// c = 0..2 (chunk 2 valid only for lane < 11 since 300-256 = 44 = 11*4).
__global__ __launch_bounds__(256) void sice_rows_kernel(
    const float* __restrict__ z_i, const float* __restrict__ z_j,
    const float* __restrict__ rate_i, const float* __restrict__ rate_j,
    float* __restrict__ partial) {
    __shared__ float probs[BLK_ROWS][NUM_C];
    __shared__ float wacc[BLK_ROWS];

    const int lane = threadIdx.x & 31;
    const int wave = threadIdx.x >> 5;
    const int row  = blockIdx.x * BLK_ROWS + wave;

    const float* __restrict__ zi = z_i + (size_t)row * NUM_C;
    const float* __restrict__ zj = z_j + (size_t)row * NUM_C;

    const int  k0   = lane * 4;
    const bool has2 = (lane < 11);
    const float NEG_INF = -__builtin_inff();

    // ---- issue both HBM streams up front (b128 per lane, fully coalesced) ----
    float zjv[12], ziv[12];
    *(float4*)&zjv[0] = *(const float4*)(zj + k0);
    *(float4*)&zjv[4] = *(const float4*)(zj + 128 + k0);
    *(float4*)&ziv[0] = *(const float4*)(zi + k0);
    *(float4*)&ziv[4] = *(const float4*)(zi + 128 + k0);
    if (has2) {
        *(float4*)&zjv[8] = *(const float4*)(zj + 256 + k0);
        *(float4*)&ziv[8] = *(const float4*)(zi + 256 + k0);
    } else {
        zjv[8] = zjv[9] = zjv[10] = zjv[11] = NEG_INF;
        ziv[8] = ziv[9] = ziv[10] = ziv[11] = NEG_INF;
    }

    // per-row shift (wave-uniform): s = -round(log(ratio)/W), round-half-even.
    const float ratio = rate_j[row] / rate_i[row];
    const int   s     = -(int)rintf(__logf(ratio) / LOG_BIN_WIDTH_F);

    // ---- softmax(z_j) -> LDS ----
    float mj = NEG_INF;
#pragma unroll
    for (int j = 0; j < 12; ++j) mj = fmaxf(mj, zjv[j]);
    mj = wave_max32(mj);

    float ej[12], sj = 0.f;
#pragma unroll
    for (int j = 0; j < 12; ++j) { ej[j] = expf(zjv[j] - mj); sj += ej[j]; }
    sj = wave_sum32(sj);
    const float rinv = 1.0f / sj;

#pragma unroll
    for (int j = 0; j < 4; ++j) probs[wave][k0 + j]       = ej[j]     * rinv;
#pragma unroll
    for (int j = 0; j < 4; ++j) probs[wave][128 + k0 + j] = ej[4 + j] * rinv;
    if (has2) {
#pragma unroll
        for (int j = 0; j < 4; ++j) probs[wave][256 + k0 + j] = ej[8 + j] * rinv;
    }

    // ---- log_softmax(z_i) in registers ----
    float mi = NEG_INF;
#pragma unroll
    for (int j = 0; j < 12; ++j) mi = fmaxf(mi, ziv[j]);
    mi = wave_max32(mi);

    float si = 0.f;
#pragma unroll
    for (int j = 0; j < 12; ++j) si += expf(ziv[j] - mi);
    si = wave_sum32(si);
    const float logd = logf(si);

    __syncthreads();   // probs[wave][*] fully written (cross-lane within wave)

    // ---- shifted dot product: mask+mod gather == (0 <= k-s < C) ----
    float acc = 0.f;
#pragma unroll
    for (int c = 0; c < 3; ++c) {
        if (c == 2 && !has2) break;
#pragma unroll
        for (int j = 0; j < 4; ++j) {
            const int k   = c * 128 + k0 + j;
            const int idx = k - s;
            if ((unsigned)idx < (unsigned)NUM_C) {
                const float lp = ziv[c * 4 + j] - mi - logd;
                acc += probs[wave][idx] * lp;
            }
        }
    }
    acc = wave_sum32(acc);
    if (lane == 0) wacc[wave] = acc;
    __syncthreads();
    if (threadIdx.x == 0) {
        float t = 0.f;
#pragma unroll
        for (int w = 0; w < BLK_ROWS; ++w) t += wacc[w];
        partial[blockIdx.x] = t;
    }
}

// Zero-fill padding slots so stage 2 needs no bounds checks.
__global__ void pad_zero_kernel(float* __restrict__ ws, int lo, int hi) {
    int i = lo + blockIdx.x * blockDim.x + threadIdx.x;
    if (i < hi) ws[i] = 0.0f;
}

// Stage 2: deterministic WMMA reduction of the (64-padded) block partials.
// A = 64 partials packed as 16x4 (any bijection works: B = ones(4x16)):
// D[m,n] += sum_k A[m,k]  -> every column of D accumulates the 16 row-sums.
// Two accumulator tiles break the WMMA C-chain RAW dependence.
__global__ __launch_bounds__(32) void sice_final_kernel(
    const float* __restrict__ partial, float* __restrict__ out,
    int n64, float scale) {
    const int lane = threadIdx.x;
    v2f bones; bones[0] = 1.0f; bones[1] = 1.0f;
    v8f d0 = {}, d1 = {};
    const float2* __restrict__ p2 = (const float2*)partial;

    int i = 0;
    for (; i + 2 <= n64; i += 2) {
        const float2 x0 = p2[(size_t)i * 32 + lane];        // global_load_b64
        const float2 x1 = p2[(size_t)(i + 1) * 32 + lane];
        v2f a0; a0[0] = x0.x; a0[1] = x0.y;
        v2f a1; a1[0] = x1.x; a1[1] = x1.y;
        d0 = __builtin_amdgcn_wmma_f32_16x16x4_f32(false, a0, false, bones,
                                                   (short)0, d0, false, false);
        d1 = __builtin_amdgcn_wmma_f32_16x16x4_f32(false, a1, false, bones,
                                                   (short)0, d1, false, false);
    }
    if (i < n64) {
        const float2 x0 = p2[(size_t)i * 32 + lane];
        v2f a0; a0[0] = x0.x; a0[1] = x0.y;
        d0 = __builtin_amdgcn_wmma_f32_16x16x4_f32(false, a0, false, bones,
                                                   (short)0, d0, false, false);
    }
    const v8f d = d0 + d1;
    // column 0 of D: lane 0 holds M=0..7, lane 16 holds M=8..15 across 8 VGPRs
    float t = d[0] + d[1] + d[2] + d[3] + d[4] + d[5] + d[6] + d[7];
    t += __shfl_xor(t, 16, 32);
    if (lane == 0) out[0] = t * scale;
}

extern "C" void kernel_launch(void* const* d_in, const int* in_sizes, int n_in,
                              void* d_out, int out_size, void* d_ws, size_t ws_size,
                              hipStream_t stream) {
    const float* z_i = (const float*)d_in[0];
    const float* z_j = (const float*)d_in[1];
    const float* ri  = (const float*)d_in[2];
    const float* rj  = (const float*)d_in[3];
    float* out = (float*)d_out;
    float* ws  = (float*)d_ws;

    const int B = in_sizes[0] / NUM_C;          // 65536
    const int nblocks = B / BLK_ROWS;           // 8192
    const int npad = (nblocks + 63) & ~63;      // multiple of 64 for WMMA tiles

    if (npad > nblocks) {
        const int extra = npad - nblocks;
        pad_zero_kernel<<<(extra + 63) / 64, 64, 0, stream>>>(ws, nblocks, npad);
    }
    sice_rows_kernel<<<nblocks, 256, 0, stream>>>(z_i, z_j, ri, rj, ws);
    sice_final_kernel<<<1, 32, 0, stream>>>(ws, out, npad / 64,
                                            -1.0f / (float)B);
}